// NT_Xent_dist_45921790328984
// MI455X (gfx1250) — compile-verified
//
#include <hip/hip_runtime.h>
#include <hip/hip_bf16.h>
#include <math.h>

typedef __attribute__((ext_vector_type(2))) float v2f;
typedef __attribute__((ext_vector_type(8))) float v8f;

#define COLS 128            // feature dim C
#define INV_T (1.0f / 0.07f)
#define SPLITS 4            // column-range splits per row block
#define LDS_STRIDE 132      // 128 + 4 pad floats -> conflict-free ds_load_b64

#if __has_builtin(__builtin_amdgcn_sched_barrier)
#define SCHED_BARRIER() __builtin_amdgcn_sched_barrier(0)
#else
#define SCHED_BARRIER() asm volatile("" ::: "memory")
#endif

// ---------------------------------------------------------------------------
// Kernel 1: L2-normalize rows of [feat1; feat2] -> Xn [2N x 128]
// One wave (32 lanes) per row; each lane owns 4 contiguous floats.
// ---------------------------------------------------------------------------
__global__ void nt_xent_normalize(const float* __restrict__ f1,
                                  const float* __restrict__ f2,
                                  float* __restrict__ Xn, int N) {
    const int row  = (blockIdx.x * blockDim.x + threadIdx.x) >> 5;
    const int lane = threadIdx.x & 31;
    const float* src = (row < N) ? (f1 + (size_t)row * COLS)
                                 : (f2 + (size_t)(row - N) * COLS);
    float4 v = *(const float4*)(src + lane * 4);
    float ss = v.x * v.x + v.y * v.y + v.z * v.z + v.w * v.w;
    ss += __shfl_xor(ss, 1);
    ss += __shfl_xor(ss, 2);
    ss += __shfl_xor(ss, 4);
    ss += __shfl_xor(ss, 8);
    ss += __shfl_xor(ss, 16);
    const float scale = 1.0f / fmaxf(sqrtf(ss), 1e-12f);
    float4 o;
    o.x = v.x * scale; o.y = v.y * scale; o.z = v.z * scale; o.w = v.w * scale;
    *(float4*)(Xn + (size_t)row * COLS + lane * 4) = o;
}

// ---------------------------------------------------------------------------
// Kernel 2: positive-pair shifted logit: pos[i] = (dot(x_i, x_{(i+N)%2N}) - 1)/T
// ---------------------------------------------------------------------------
__global__ void nt_xent_pos(const float* __restrict__ Xn,
                            float* __restrict__ pos, int two_n, int N) {
    const int row  = (blockIdx.x * blockDim.x + threadIdx.x) >> 5;
    const int lane = threadIdx.x & 31;
    const int prt  = (row + N) % two_n;
    float4 a = *(const float4*)(Xn + (size_t)row * COLS + lane * 4);
    float4 b = *(const float4*)(Xn + (size_t)prt * COLS + lane * 4);
    float d = a.x * b.x + a.y * b.y + a.z * b.z + a.w * b.w;
    d += __shfl_xor(d, 1);
    d += __shfl_xor(d, 2);
    d += __shfl_xor(d, 4);
    d += __shfl_xor(d, 8);
    d += __shfl_xor(d, 16);
    if (lane == 0) pos[row] = (d - 1.0f) * INV_T;
}

// ---------------------------------------------------------------------------
// Kernel 3: WMMA row statistics with LDS-staged B tiles.
// Block = 256 threads = 8 waves; block owns 128 rows (wave w -> 16-row strip)
// and one of SPLITS column ranges. Per column tile (16 cols x 128 K):
//   - all 256 threads cooperatively stage the 8KB B tile into padded LDS
//   - each wave reads all 32 B fragments from LDS (conflict-free), then a
//     scheduling barrier forces a single s_wait_dscnt before an uninterrupted
//     burst of 32x V_WMMA_F32_16X16X4_F32 against resident A fragments
//   - exp((c-1)/T) accumulated per row, diagonal masked
// Next tile's global data is prefetched into registers before the consume
// phase so L2 latency overlaps the WMMA work.
//
// f32 WMMA fragment layouts (ISA 7.12.2):
//   A 16x4:  lane&15 = M, VGPR{0,1} = K{0,1} (lanes 0-15) / K{2,3} (16-31)
//   B 4x16:  lane&15 = N, same K striping
//   C 16x16: lane&15 = N, VGPR v -> M = v + 8*(lane>=16)
// ---------------------------------------------------------------------------
__global__ void __launch_bounds__(256)
nt_xent_rowstats(const float* __restrict__ Xn, float* __restrict__ Spart,
                 int two_n, int colTilesPerSplit) {
    __shared__ float Btile[16 * LDS_STRIDE];

    const int tid    = threadIdx.x;
    const int wave   = tid >> 5;
    const int lane   = tid & 31;
    const int split  = blockIdx.x % SPLITS;       // column range
    const int rowBlk = blockIdx.x / SPLITS;       // 128-row block
    const int r0     = rowBlk * 128 + wave * 16;  // this wave's 16-row strip
    const int half   = lane >> 4;                 // K sub-pair select
    const int m      = lane & 15;                 // M (A) / N (B/C)

    // Resident A fragments: 32 K-blocks of v2f for this wave's 16 rows.
    const float* arow = Xn + (size_t)(r0 + m) * COLS + 2 * half;
    v2f afrag[32];
#pragma unroll
    for (int kb = 0; kb < 32; ++kb)
        afrag[kb] = *(const v2f*)(arow + kb * 4);

    float rs[8];
#pragma unroll
    for (int v = 0; v < 8; ++v) rs[v] = 0.0f;

    const int ctBeg = split * colTilesPerSplit;
    const int ctEnd = ctBeg + colTilesPerSplit;

    // Staging indices: 512 float4 per tile, 2 per thread.
    const int f4a = tid;            // float4 index 0..255
    const int f4b = tid + 256;      // float4 index 256..511
    const int rowA = f4a >> 5, colA = (f4a & 31) * 4;
    const int rowB = f4b >> 5, colB = (f4b & 31) * 4;

    // Prefetch first tile's global data into registers.
    const float* gbase = Xn + (size_t)ctBeg * 16 * COLS;
    float4 pfA = *(const float4*)(gbase + f4a * 4);
    float4 pfB = *(const float4*)(gbase + f4b * 4);

    for (int ct = ctBeg; ct < ctEnd; ++ct) {
        __syncthreads();   // previous tile fully consumed
        *(float4*)(&Btile[rowA * LDS_STRIDE + colA]) = pfA;
        *(float4*)(&Btile[rowB * LDS_STRIDE + colB]) = pfB;
        __syncthreads();   // tile visible to all waves

        // Prefetch next tile while consuming this one.
        if (ct + 1 < ctEnd) {
            const float* gnext = Xn + (size_t)(ct + 1) * 16 * COLS;
            pfA = *(const float4*)(gnext + f4a * 4);
            pfB = *(const float4*)(gnext + f4b * 4);
        }

        // Phase 1: issue ALL 32 B-fragment LDS reads (paired ds_load_2addr),
        // pipelined back-to-back; padded stride keeps them bank-conflict-free.
        const float* bbase = &Btile[m * LDS_STRIDE + 2 * half];
        v2f bfrag[32];
#pragma unroll
        for (int kb = 0; kb < 32; ++kb)
            bfrag[kb] = *(const v2f*)(bbase + kb * 4);

        // Keep loads and WMMAs in separate phases: one s_wait_dscnt total.
        SCHED_BARRIER();

        // Phase 2: uninterrupted WMMA burst (D->C accumulation chain is
        // back-to-back legal per ISA 7.12.1 hazard table).
        v8f c = {};
#pragma unroll
        for (int kb = 0; kb < 32; ++kb) {
            c = __builtin_amdgcn_wmma_f32_16x16x4_f32(
                    /*neg_a=*/false, afrag[kb],
                    /*neg_b=*/false, bfrag[kb],
                    /*c_mod=*/(short)0, c,
                    /*reuse_a=*/false, /*reuse_b=*/false);
        }

        SCHED_BARRIER();

        // exp((cos - 1)/T), masked diagonal; row i = r0+v+8*half, col j = ct*16+m
        const int jcol = ct * 16 + m;
#pragma unroll
        for (int v = 0; v < 8; ++v) {
            const int irow = r0 + v + 8 * half;
            const float e  = __expf((c[v] - 1.0f) * INV_T);
            rs[v] += (irow == jcol) ? 0.0f : e;
        }
    }

    // Reduce each row sum across its 16-lane half; lanes 0 / 16 write.
#pragma unroll
    for (int v = 0; v < 8; ++v) {
        float s = rs[v];
        s += __shfl_xor(s, 1);
        s += __shfl_xor(s, 2);
        s += __shfl_xor(s, 4);
        s += __shfl_xor(s, 8);
        if (m == 0)
            Spart[(size_t)split * two_n + r0 + v + 8 * half] = s;
    }
}

// ---------------------------------------------------------------------------
// Kernel 4: loss = -(1/2N) * sum_i ( pos[i] - log( sum_splits Spart[s][i] ) )
// ---------------------------------------------------------------------------
__global__ void nt_xent_finalize(const float* __restrict__ Spart,
                                 const float* __restrict__ pos,
                                 float* __restrict__ out, int two_n) {
    __shared__ float red[256];
    float acc = 0.0f;
    for (int i = threadIdx.x; i < two_n; i += 256) {
        float s = Spart[i] + Spart[two_n + i] + Spart[2 * two_n + i] +
                  Spart[3 * two_n + i];
        acc += pos[i] - logf(s);
    }
    red[threadIdx.x] = acc;
    __syncthreads();
    for (int off = 128; off > 0; off >>= 1) {
        if (threadIdx.x < off) red[threadIdx.x] += red[threadIdx.x + off];
        __syncthreads();
    }
    if (threadIdx.x == 0)
        out[0] = -red[0] / (float)two_n;   // TEMPERATURE / BASE_TEMPERATURE == 1
}

// ---------------------------------------------------------------------------
extern "C" void kernel_launch(void* const* d_in, const int* in_sizes, int n_in,
                              void* d_out, int out_size, void* d_ws, size_t ws_size,
                              hipStream_t stream) {
    const float* f1 = (const float*)d_in[0];
    const float* f2 = (const float*)d_in[1];
    float* out = (float*)d_out;

    const int N     = in_sizes[0] / COLS;   // 4096
    const int two_n = 2 * N;                // 8192

    // Workspace layout (floats)
    float* Xn    = (float*)d_ws;                       // two_n * 128
    float* Spart = Xn + (size_t)two_n * COLS;          // SPLITS * two_n
    float* pos   = Spart + (size_t)SPLITS * two_n;     // two_n

    // 1 wave per row for normalize / pos kernels: 8 rows per 256-thr block.
    const int rowBlocks = two_n / 8;
    nt_xent_normalize<<<rowBlocks, 256, 0, stream>>>(f1, f2, Xn, N);
    nt_xent_pos<<<rowBlocks, 256, 0, stream>>>(Xn, pos, two_n, N);

    // WMMA stats: (two_n/128) row blocks * SPLITS column ranges.
    const int rowBlkCnt        = two_n / 128;                  // 64
    const int colTilesPerSplit = (two_n / 16) / SPLITS;        // 128
    const int statBlocks       = rowBlkCnt * SPLITS;           // 256
    nt_xent_rowstats<<<statBlocks, 256, 0, stream>>>(Xn, Spart, two_n,
                                                     colTilesPerSplit);

    nt_xent_finalize<<<1, 256, 0, stream>>>(Spart, pos, out, two_n);
}